// S4_28381143892000
// MI455X (gfx1250) — compile-verified
//
#include <hip/hip_runtime.h>
#include <math.h>

// S4 layer for MI455X (gfx1250, wave32).
// Stages (all L2-resident, ~28MB workspace; HBM floor ~40MB => ~1.7us @23.3TB/s):
//   0: transpose u[B,L,H] -> ut[B,H,L]        (coalesced LDS tile transpose)
//   1: Cauchy kernel at roots of unity -> at_roots[H,L] (fp32 VALU, ~1.6 GFLOP)
//   2: inverse FFT (radix-2 DIT in LDS, LDS twiddle table) -> k[H,L] real
//   3: causal conv as block-Toeplitz matmul on V_WMMA_F32_16X16X4_F32 (~17 GFLOP),
//      k-row staged into LDS by the Tensor Data Mover (tensor_load_to_lds).

#define HCH   512
#define NST   64
#define LSEQ  2048
#define NBATCH 4

typedef __attribute__((ext_vector_type(2))) float v2f;
typedef __attribute__((ext_vector_type(8))) float v8f;
typedef __attribute__((ext_vector_type(4))) unsigned v4u;
typedef __attribute__((ext_vector_type(8))) unsigned v8u;

// ---------------- Stage 0: tiled transpose u[B,L,H] -> ut[B,H,L] --------------
__global__ __launch_bounds__(256) void s4_transpose(const float* __restrict__ u,
                                                    float* __restrict__ ut) {
  __shared__ float tile[32][33];
  const int b   = blockIdx.z;
  const int lt0 = blockIdx.x * 32;
  const int h0  = blockIdx.y * 32;
  const int tx = threadIdx.x, ty = threadIdx.y;
  const float* ub = u + (size_t)b * LSEQ * HCH;
  for (int j = ty; j < 32; j += 8)
    tile[j][tx] = ub[(size_t)(lt0 + j) * HCH + h0 + tx];
  __syncthreads();
  float* utb = ut + (size_t)b * HCH * LSEQ;
  for (int j = ty; j < 32; j += 8)
    utb[(size_t)(h0 + j) * LSEQ + lt0 + tx] = tile[tx][j];
}

// ---------------- Stage 1: Cauchy kernel at the roots of unity ----------------
// at_roots[h,l] = (1/L) * toeplitz(l) * (k00 - k01 * k10 / (1 + k11))
// One complex reciprocal per (h,l,n) shared by all four numerators.
__global__ __launch_bounds__(256) void s4_cauchy(
    const float* __restrict__ lamr_g, const float* __restrict__ lami_g,
    const float* __restrict__ Pg, const float* __restrict__ Bg,
    const float* __restrict__ logstep,
    const float* __restrict__ Crg, const float* __restrict__ Cig,
    float* __restrict__ atr_re, float* __restrict__ atr_im) {
  __shared__ float s_lr[NST], s_li[NST], s_cbr[NST], s_cbi[NST],
                   s_cpr[NST], s_cpi[NST], s_pb[NST], s_pp[NST];
  const int h = blockIdx.x >> 3;
  const int l = ((blockIdx.x & 7) << 8) + threadIdx.x;
  if (threadIdx.x < NST) {
    const int n   = threadIdx.x;
    const int idx = h * NST + n;
    const float lr = lamr_g[idx], li = lami_g[idx];
    const float p  = Pg[idx],     bb = Bg[idx];
    const float cr = Crg[idx],    ci = Cig[idx];
    s_lr[n]  = fmaxf(lr, -1e-4f);
    s_li[n]  = li;
    s_cbr[n] = cr * bb;  s_cbi[n] = ci * bb;   // C * B
    s_cpr[n] = cr * p;   s_cpi[n] = ci * p;    // C * P
    s_pb[n]  = p * bb;                          // P * B (real)
    s_pp[n]  = p * p;                           // P * P (real)
  }
  __syncthreads();

  const float step  = expf(logstep[h]);
  const float theta = 6.28318530717958647692f * (float)l * (1.0f / (float)LSEQ);
  const float wr = cosf(theta);
  const float wi = -sinf(theta);                // omega = exp(-2*pi*i*l/L)

  // g = (2/step) * (1-omega)/(1+omega)
  const float vr = 1.0f + wr, vi = wi;
  const float inv_v = 1.0f / (vr * vr + vi * vi);
  const float ur = 1.0f - wr, ui = -wi;
  const float gs = 2.0f / step;
  const float g_r = gs * (ur * vr + ui * vi) * inv_v;
  const float g_i = gs * (ui * vr - ur * vi) * inv_v;

  float k00r = 0.f, k00i = 0.f, k01r = 0.f, k01i = 0.f;
  float k10r = 0.f, k10i = 0.f, k11r = 0.f, k11i = 0.f;
#pragma unroll 8
  for (int n = 0; n < NST; ++n) {
    const float dr = g_r - s_lr[n];
    const float di = g_i - s_li[n];
    const float s  = 1.0f / (dr * dr + di * di);
    const float ir = dr * s;                    // Re 1/(g-Lam)
    const float im = -di * s;                   // Im 1/(g-Lam)
    k00r += s_cbr[n] * ir - s_cbi[n] * im;
    k00i += s_cbr[n] * im + s_cbi[n] * ir;
    k01r += s_cpr[n] * ir - s_cpi[n] * im;
    k01i += s_cpr[n] * im + s_cpi[n] * ir;
    k10r += s_pb[n] * ir;  k10i += s_pb[n] * im;
    k11r += s_pp[n] * ir;  k11i += s_pp[n] * im;
  }

  const float denr = 1.0f + k11r, deni = k11i;
  const float sd   = 1.0f / (denr * denr + deni * deni);
  const float t_r  = denr * sd, t_i = -deni * sd;          // 1/(1+k11)
  const float q_r  = k01r * t_r - k01i * t_i;
  const float q_i  = k01r * t_i + k01i * t_r;
  const float q2r  = q_r * k10r - q_i * k10i;              // k01*t*k10
  const float q2i  = q_r * k10i + q_i * k10r;
  const float ar_r = k00r - q2r;
  const float ar_i = k00i - q2i;
  const float tp_r = 2.0f * vr * inv_v;                    // 2/(1+omega)
  const float tp_i = -2.0f * vi * inv_v;
  const float sc   = 1.0f / (float)LSEQ;                   // fold ifft 1/L
  atr_re[(size_t)h * LSEQ + l] = (tp_r * ar_r - tp_i * ar_i) * sc;
  atr_im[(size_t)h * LSEQ + l] = (tp_r * ar_i + tp_i * ar_r) * sc;
}

// ---------------- Stage 2: per-channel 2048-pt inverse FFT in LDS -------------
__global__ __launch_bounds__(256) void s4_ifft(const float* __restrict__ atr_re,
                                               const float* __restrict__ atr_im,
                                               float* __restrict__ kker) {
  __shared__ float re[LSEQ], im[LSEQ];
  __shared__ float twr_t[LSEQ / 2], twi_t[LSEQ / 2];
  const int h = blockIdx.x;
  const float* ar = atr_re + (size_t)h * LSEQ;
  const float* ai = atr_im + (size_t)h * LSEQ;
  // twiddle table: exp(+2*pi*i*j/L), j < L/2  (inverse transform sign)
  for (int j = threadIdx.x; j < (LSEQ / 2); j += 256) {
    const float ang = 6.28318530717958647692f * (float)j * (1.0f / (float)LSEQ);
    twr_t[j] = cosf(ang);
    twi_t[j] = sinf(ang);
  }
  // bit-reversed load (11 bits)
  for (int t = threadIdx.x; t < LSEQ; t += 256) {
    const unsigned rev = __brev((unsigned)t) >> 21;
    re[t] = ar[rev];
    im[t] = ai[rev];
  }
  __syncthreads();
  for (int s = 1; s <= 11; ++s) {
    const int half = 1 << (s - 1);
    for (int j = threadIdx.x; j < (LSEQ >> 1); j += 256) {
      const int grp = j >> (s - 1);
      const int pos = j & (half - 1);
      const int i0  = (grp << s) + pos;
      const int i1  = i0 + half;
      const int tj  = pos << (11 - s);          // pos/m in units of 1/L
      const float twr = twr_t[tj], twi = twi_t[tj];
      const float tr = twr * re[i1] - twi * im[i1];
      const float ti = twr * im[i1] + twi * re[i1];
      re[i1] = re[i0] - tr; im[i1] = im[i0] - ti;
      re[i0] += tr;         im[i0] += ti;
    }
    __syncthreads();
  }
  for (int t = threadIdx.x; t < LSEQ; t += 256)
    kker[(size_t)h * LSEQ + t] = re[t];   // real part; 1/L folded in stage 1
}

// ---------------- Stage 3: causal conv as block-Toeplitz WMMA f32 -------------
// out[b,h,16*i+m] = D[h]*u + sum_d sum_kk K_h(d)[m,kk] * ut[b,h,16*(i-d)+kk]
// k_h staged into LDS by the Tensor Data Mover (one DMA per block).
__global__ __launch_bounds__(256) void s4_conv_wmma(
    const float* __restrict__ kker, const float* __restrict__ ut,
    const float* __restrict__ Dg, float* __restrict__ out) {
  __shared__ float ks[LSEQ];
  const int h = blockIdx.x >> 2;
  const int q = blockIdx.x & 3;

  // ---- TDM: DMA the contiguous 8KB row k[h,:] into LDS (wave 0 issues). ----
  if ((threadIdx.x >> 5) == 0) {
    const unsigned long long gaddr =
        (unsigned long long)(const void*)(kker + (size_t)h * LSEQ);
    const unsigned ldsoff = (unsigned)(unsigned long long)(void*)ks;
    // D# group0: count=1, lds_addr, global_addr[56:0], type=2 (ISA 8.3)
    v4u g0;
    g0.x = 1u;
    g0.y = ldsoff;
    g0.z = (unsigned)(gaddr & 0xFFFFFFFFull);
    g0.w = (unsigned)((gaddr >> 32) & 0x01FFFFFFull) | (2u << 30);
    // D# group1: wg_mask=0 (not in cluster), data_size=2 (4B),
    // tensor_dim0=2048, tensor_dim1=1, tile_dim0=2048, tile_dim1=1,
    // tensor_dim0_stride=2048 (ISA 8.4)
    v8u g1;
    g1.s0 = 0x00020000u;                       // data_size=4B
    g1.s1 = (LSEQ & 0xFFFFu) << 16;            // tensor_dim0[15:0] @ bit48
    g1.s2 = (LSEQ >> 16) | (1u << 16);         // tensor_dim0[31:16] | tensor_dim1 lo
    g1.s3 = ((unsigned)LSEQ) << 16;            // tensor_dim1 hi=0 | tile_dim0
    g1.s4 = 1u;                                // tile_dim1=1, tile_dim2=0
    g1.s5 = (unsigned)LSEQ;                    // tensor_dim0_stride[31:0]
    g1.s6 = 0u;                                // stride0 hi | stride1 lo
    g1.s7 = 0u;                                // stride1 hi
    asm volatile("tensor_load_to_lds %0, %1" :: "s"(g0), "s"(g1) : "memory");
#if __has_builtin(__builtin_amdgcn_s_wait_tensorcnt)
    __builtin_amdgcn_s_wait_tensorcnt(0);
#else
    asm volatile("s_wait_tensorcnt 0x0" ::: "memory");
#endif
  }
  __syncthreads();

  const int wave = threadIdx.x >> 5;           // 8 waves per block (wave32)
  const int lane = threadIdx.x & 31;
  const int ct   = q * 8 + wave;               // column-tile 0..31
  const int i0   = ct * 4;                     // base output-block index
  const int mrow = lane & 15;                  // A-matrix row (M)
  const int hi   = lane >> 4;                  // lane-half selects K {0,1}/{2,3}
  const int col  = lane & 15;                  // B/C/D column (N)
  const int bN   = col >> 2;                   // batch of this column
  const int iN   = i0 + (col & 3);             // output block of this column

  const float* utb = ut + (size_t)bN * HCH * LSEQ + (size_t)h * LSEQ;
  v8f c = {};                                   // fp32 accumulator tile

  // ---- d = 0: diagonal (lower-triangular) block, A needs zero masking ----
  {
    const int tb = iN * 16;                     // always >= 0
#pragma unroll
    for (int kb = 0; kb < 16; kb += 4) {
      const int kkA = kb + hi * 2;
      const int ia0 = mrow - kkA;
      const int ia1 = ia0 - 1;
      v2f a, bvec;
      a.x = (ia0 >= 0) ? ks[ia0] : 0.0f;
      a.y = (ia1 >= 0) ? ks[ia1] : 0.0f;
      bvec.x = utb[tb + kkA];
      bvec.y = utb[tb + kkA + 1];
      c = __builtin_amdgcn_wmma_f32_16x16x4_f32(false, a, false, bvec,
                                                (short)0, c, false, false);
    }
  }
  // ---- d >= 1: full Toeplitz blocks, in-range A gathers, masked B columns ----
#pragma unroll 2
  for (int d = 1; d <= i0 + 3; ++d) {
    const int  tbase = (iN - d) * 16;           // causal: negative => zeros
    const bool valid = (tbase >= 0);
    const int  tb    = valid ? tbase : 0;
    __builtin_prefetch(utb + tb, 0, 1);         // global_prefetch of streamed u
#pragma unroll
    for (int kb = 0; kb < 16; kb += 4) {
      const int kkA = kb + hi * 2;
      const int ia0 = 16 * d + mrow - kkA;      // >= 1, in range for d >= 1
      v2f a, bvec;
      a.x = ks[ia0];
      a.y = ks[ia0 - 1];
      const float b0 = utb[tb + kkA];
      const float b1 = utb[tb + kkA + 1];
      bvec.x = valid ? b0 : 0.0f;
      bvec.y = valid ? b1 : 0.0f;
      c = __builtin_amdgcn_wmma_f32_16x16x4_f32(false, a, false, bvec,
                                                (short)0, c, false, false);
    }
  }

  const float dh = Dg[h];
#pragma unroll
  for (int r = 0; r < 8; ++r) {
    const int M = r + hi * 8;                   // C layout: VGPR r holds M / M+8
    const int t = iN * 16 + M;
    const float uval = utb[t];
    out[(size_t)bN * LSEQ * HCH + (size_t)t * HCH + h] = c[r] + dh * uval;
  }
}

// ------------------------------- launcher ------------------------------------
extern "C" void kernel_launch(void* const* d_in, const int* in_sizes, int n_in,
                              void* d_out, int out_size, void* d_ws, size_t ws_size,
                              hipStream_t stream) {
  const float* u      = (const float*)d_in[0];
  const float* lam_re = (const float*)d_in[1];
  const float* lam_im = (const float*)d_in[2];
  const float* P      = (const float*)d_in[3];
  const float* B      = (const float*)d_in[4];
  const float* D      = (const float*)d_in[5];
  const float* lstep  = (const float*)d_in[6];
  const float* C_re   = (const float*)d_in[7];
  const float* C_im   = (const float*)d_in[8];
  float* out = (float*)d_out;

  // workspace: atr_re | atr_im | k | ut   (~28 MB, fits L2 with room to spare)
  float* w      = (float*)d_ws;
  float* atr_re = w;
  float* atr_im = w + (size_t)HCH * LSEQ;
  float* kker   = w + 2 * (size_t)HCH * LSEQ;
  float* ut     = w + 3 * (size_t)HCH * LSEQ;

  dim3 tgrid(LSEQ / 32, HCH / 32, NBATCH);
  s4_transpose<<<tgrid, dim3(32, 8), 0, stream>>>(u, ut);

  s4_cauchy<<<HCH * (LSEQ / 256), 256, 0, stream>>>(
      lam_re, lam_im, P, B, lstep, C_re, C_im, atr_re, atr_im);

  s4_ifft<<<HCH, 256, 0, stream>>>(atr_re, atr_im, kker);

  s4_conv_wmma<<<HCH * 4, 256, 0, stream>>>(kker, ut, D, out);
}